// GammaNorm2D_18829136625727
// MI455X (gfx1250) — compile-verified
//
#include <hip/hip_runtime.h>
#include <cmath>

// Problem constants (fixed by the reference: x is (256, 224, 224) f32)
#define N_PIX  50176   // 224*224
#define NV4    (N_PIX / 4)   // 12544 float4 chunks
#define N_ROWS 256
#define TPB    1024    // 32 wave32 waves per workgroup
#define NWAVES (TPB / 32)

typedef float v2f __attribute__((ext_vector_type(2)));
typedef float v8f __attribute__((ext_vector_type(8)));
// Matches the async builtin's expected pointee: 'int __attribute__((vector_size(16)))'
typedef int v4i_b __attribute__((vector_size(16)));

#if __has_builtin(__builtin_amdgcn_wmma_f32_16x16x4_f32)
#define HAVE_WMMA_F32X4 1
#else
#define HAVE_WMMA_F32X4 0
#endif

#if __has_builtin(__builtin_amdgcn_global_load_async_to_lds_b128)
#define HAVE_ASYNC_LDS 1
#else
#define HAVE_ASYNC_LDS 0
#endif

// ---------------------------------------------------------------------------
// Wave-level sum of 32 lane values using V_WMMA_F32_16X16X4_F32 as a
// horizontal-add engine. A (16x4 f32, 2 VGPRs/lane): A[m][0]=v(lane m),
// A[m][2]=v(lane m+16), A[m][1]=A[m][3]=0. B = ones. Then
// D[m][n] = v[m] + v[m+16] for every column n. Sum the 8 D VGPRs (rows 0-7 in
// lanes 0-15, rows 8-15 in lanes 16-31), and one xor-16 shuffle finishes.
// Requires EXEC all-ones in the wave: call sites are wave-uniform.
// ---------------------------------------------------------------------------
__device__ __forceinline__ float wave_sum32(float v) {
#if HAVE_WMMA_F32X4
  v2f a; a[0] = v;    a[1] = 0.0f;
  v2f b; b[0] = 1.0f; b[1] = 1.0f;
  v8f c = {0.0f, 0.0f, 0.0f, 0.0f, 0.0f, 0.0f, 0.0f, 0.0f};
  c = __builtin_amdgcn_wmma_f32_16x16x4_f32(false, a, false, b, (short)0, c,
                                            false, false);
  float s = ((c[0] + c[1]) + (c[2] + c[3])) + ((c[4] + c[5]) + (c[6] + c[7]));
  s += __shfl_xor(s, 16, 32);
  return s;
#else
  for (int off = 16; off > 0; off >>= 1) v += __shfl_xor(v, off, 32);
  return v;
#endif
}

__device__ __forceinline__ float wave_min32(float v) {
  for (int off = 16; off > 0; off >>= 1) v = fminf(v, __shfl_xor(v, off, 32));
  return v;
}

// ---------------------------------------------------------------------------
// Scalar special functions (f32, executed by one thread per row)
// ---------------------------------------------------------------------------
__device__ __forceinline__ float dev_gamma(float z) {
  const float CH[8] = {676.5203681218851f,    -1259.1392167224028f,
                       771.3234287776531f,    -176.6150291621406f,
                       12.507343278686905f,   -0.13857109526572012f,
                       9.984369578019572e-06f, 1.5056327351493116e-07f};
  float acc = 0.9999999999998099f;
#pragma unroll
  for (int i = 0; i < 8; ++i) acc += CH[i] / (z + (float)(i + 1));
  float t = z + 7.5f;  // z + 8 - 0.5
  float y = 2.5066282746310002f * expf((z + 0.5f) * logf(t)) * expf(-t) * acc;
  return y / z;
}

__device__ __forceinline__ float dev_digamma(float x) {
  // Recurrence up to x >= 6, then asymptotic series.
  float r = 0.0f;
  while (x < 6.0f) { r -= 1.0f / x; x += 1.0f; }
  float f = 1.0f / (x * x);
  float psi = logf(x) - 0.5f / x
            - f * (0.08333333333333333f
                   - f * (0.008333333333333333f - f * 0.003968253968253968f));
  return psi + r;
}

__device__ __forceinline__ float dev_trigamma(float x) {  // reference formula
  float z = x + 1.0f;
  float zz = z * z;
  float a = 0.2f - 1.0f / (7.0f * zz);
  float b = 1.0f - a / zz;
  float c = 1.0f + b / (3.0f * z);
  float d = 1.0f + c / (2.0f * z);
  return d / z + 1.0f / (x * x);
}

// ---------------------------------------------------------------------------
// One workgroup per batch row. Phases:
//   0) async-stage the 200KB row into LDS (CDNA5: 320KB LDS / WGP)
//   1) row min (float4 LDS reads, shuffle + LDS reduction)
//   2) sum(v) and sum(log v), v = x - min + 2e-7  (float4 + WMMA reductions)
//   3) thread 0: Newton solve for k, theta; precompute series coeffs
//   4) per-pixel regularized lower incomplete gamma from LDS, b128 stores
// ---------------------------------------------------------------------------
__global__ __launch_bounds__(TPB, 1) void gamma_norm_kernel(
    const float* __restrict__ x, float* __restrict__ out) {
  extern __shared__ float smem[];
  float* srow  = smem;                 // N_PIX floats: staged row
  float* sredA = smem + N_PIX;         // NWAVES partials
  float* sredB = sredA + NWAVES;       // NWAVES partials
  float* sprm  = sredB + NWAVES;       // 16 floats of row parameters

  const int tid  = (int)threadIdx.x;
  const int lane = tid & 31;
  const int wid  = tid >> 5;
  const int row  = (int)blockIdx.x;
  const float* g = x + (size_t)row * N_PIX;
  const float4* srow4 = reinterpret_cast<const float4*>(srow);

  // ---- Phase 0: stage row into LDS (async DMA, ASYNCcnt) ----------------
  for (int i = tid; i < NV4; i += TPB) {
#if HAVE_ASYNC_LDS
    __builtin_amdgcn_global_load_async_to_lds_b128(
        (v4i_b*)(g + 4 * i),
        (v4i_b*)(srow + 4 * i),
        /*offset=*/0, /*cpol=*/0);
#else
    reinterpret_cast<float4*>(srow)[i] =
        reinterpret_cast<const float4*>(g)[i];
#endif
  }
#if HAVE_ASYNC_LDS
#if __has_builtin(__builtin_amdgcn_s_wait_asynccnt)
  __builtin_amdgcn_s_wait_asynccnt(0);
#else
  asm volatile("s_wait_asynccnt 0x0" ::: "memory");
#endif
#endif
  __syncthreads();

  // ---- Phase 1: row minimum (4-wide) ------------------------------------
  const float FMAX = 3.402823466e+38f;
  float4 mn4 = {FMAX, FMAX, FMAX, FMAX};
  for (int i = tid; i < NV4; i += TPB) {
    float4 v = srow4[i];
    mn4.x = fminf(mn4.x, v.x);
    mn4.y = fminf(mn4.y, v.y);
    mn4.z = fminf(mn4.z, v.z);
    mn4.w = fminf(mn4.w, v.w);
  }
  float mn = fminf(fminf(mn4.x, mn4.y), fminf(mn4.z, mn4.w));
  mn = wave_min32(mn);
  if (lane == 0) sredA[wid] = mn;
  __syncthreads();
  if (wid == 0) {
    float a = wave_min32(sredA[lane]);
    if (lane == 0) sprm[0] = a;
  }
  __syncthreads();
  const float minv = sprm[0];

  // ---- Phase 2: sum and sum-of-log (shift +2e-7 per reference) ----------
  const float sub2 = minv - 2e-7f;  // v = x - min + 2e-7
  float4 aS = {0.0f, 0.0f, 0.0f, 0.0f};
  float4 aL = {0.0f, 0.0f, 0.0f, 0.0f};
  for (int i = tid; i < NV4; i += TPB) {
    float4 v = srow4[i];
    float v0 = v.x - sub2, v1 = v.y - sub2, v2 = v.z - sub2, v3 = v.w - sub2;
    aS.x += v0; aS.y += v1; aS.z += v2; aS.w += v3;
    aL.x += logf(v0); aL.y += logf(v1); aL.z += logf(v2); aL.w += logf(v3);
  }
  float accS = (aS.x + aS.y) + (aS.z + aS.w);
  float accL = (aL.x + aL.y) + (aL.z + aL.w);
  float ws = wave_sum32(accS);
  float wl = wave_sum32(accL);
  if (lane == 0) { sredA[wid] = ws; sredB[wid] = wl; }
  __syncthreads();

  // ---- Phase 3: per-row scalar solve (wave 0) ---------------------------
  if (wid == 0) {
    float a = wave_sum32(sredA[lane]);
    float b = wave_sum32(sredB[lane]);
    if (lane == 0) {
      const float Ninv = 1.0f / (float)N_PIX;
      float mean    = a * Ninv;
      float meanlog = b * Ninv;
      float s  = logf(mean) - meanlog;           // s > 0 by Jensen
      float s3 = s - 3.0f;
      float rt = sqrtf(s3 * s3 + 24.0f * s);
      float k  = (3.0f - s + rt) / (12.0f * s) + 1e-7f;
#pragma unroll 1
      for (int it = 0; it < 10; ++it) {          // Newton on log k - psi(k) = s
        float nm = logf(k) - dev_digamma(k) - s;
        float dn = 1.0f / k - dev_trigamma(k);
        k = k - nm / dn;
      }
      k = fminf(fmaxf(k, 1e-7f), 18.0f);
      float th = mean / k;
      float gk = dev_gamma(k);
      sprm[1] = k;
      sprm[2] = 1.0f / th;
      sprm[3] = gk;
      sprm[4] = 1.0f / gk;
      // Series coeffs c_j = 1/Gamma(k+j+1) built exactly as the reference:
      float gs = gk * k;                         // Gamma(k+1)
      sprm[5] = 1.0f / gs;
#pragma unroll 1
      for (int j = 1; j <= 7; ++j) { gs *= (k + (float)j); sprm[5 + j] = 1.0f / gs; }
    }
  }
  __syncthreads();

  const float k      = sprm[1];
  const float inv_th = sprm[2];
  const float gk     = sprm[3];
  const float rgk    = sprm[4];
  const float c0 = sprm[5],  c1 = sprm[6],  c2 = sprm[7],  c3 = sprm[8];
  const float c4 = sprm[9],  c5 = sprm[10], c6 = sprm[11], c7 = sprm[12];
  const float sub1 = minv - 1e-7f;  // u = (x - min + 1e-7) / theta

  // ---- Phase 4: per-pixel lower incomplete gamma (4-wide) ---------------
  float4* o4 = reinterpret_cast<float4*>(out + (size_t)row * N_PIX);
  for (int i = tid; i < NV4; i += TPB) {
    float4 v = srow4[i];
    float4 r;
#pragma unroll
    for (int c = 0; c < 4; ++c) {
      float xv = (c == 0) ? v.x : (c == 1) ? v.y : (c == 2) ? v.z : v.w;
      float u = (xv - sub1) * inv_th;             // u > 0
      // R = sum_j c_j u^j  (Horner, 7 FMAs)
      float R = c7;
      R = fmaf(R, u, c6); R = fmaf(R, u, c5); R = fmaf(R, u, c4);
      R = fmaf(R, u, c3); R = fmaf(R, u, c2); R = fmaf(R, u, c1);
      R = fmaf(R, u, c0);
      // L = u^k * Gamma(k) * e^{-u}; y = (1/Gamma(k)) * L * R
      float L = expf(fmaf(k, logf(u), -u)) * gk;
      float y = rgk * L * R;
      y = isfinite(y) ? y : 0.0f;
      if (c == 0) r.x = y; else if (c == 1) r.y = y;
      else if (c == 2) r.z = y; else r.w = y;
    }
    o4[i] = r;
  }
}

// ---------------------------------------------------------------------------
extern "C" void kernel_launch(void* const* d_in, const int* in_sizes, int n_in,
                              void* d_out, int out_size, void* d_ws,
                              size_t ws_size, hipStream_t stream) {
  (void)in_sizes; (void)n_in; (void)out_size; (void)d_ws; (void)ws_size;
  const float* x = (const float*)d_in[0];
  float* out = (float*)d_out;

  const size_t shmem = (size_t)(N_PIX + 2 * NWAVES + 16) * sizeof(float);
  // CDNA5 allows up to 320KB LDS per workgroup; opt in past the default cap.
  (void)hipFuncSetAttribute((const void*)gamma_norm_kernel,
                            hipFuncAttributeMaxDynamicSharedMemorySize,
                            (int)shmem);
  gamma_norm_kernel<<<N_ROWS, TPB, shmem, stream>>>(x, out);
}